// RochetNet_16063177687589
// MI455X (gfx1250) — compile-verified
//
#include <hip/hip_runtime.h>
#include <float.h>

typedef __attribute__((ext_vector_type(2))) float v2f;
typedef __attribute__((ext_vector_type(4))) float v4f;
typedef __attribute__((ext_vector_type(8))) float v8f;

#define NITEMS 64
#define NHID 256
#define ROWS_PER_BLOCK 128
#define SVPITCH 66   // pad 64->66 floats: A-frag row-strided ds_load_b64 avoids bank conflicts

__global__ __launch_bounds__(256) void rochet_wmma_kernel(
    const float* __restrict__ val,   // [B, 64]
    const float* __restrict__ W,     // [256, 64]
    const float* __restrict__ w0,    // [256]
    float* __restrict__ alloc_out,   // [B, 64]
    float* __restrict__ pay_out,     // [B]
    int batch)
{
    __shared__ float sV[ROWS_PER_BLOCK * SVPITCH];  // ~33 KB
    __shared__ float sw0[NHID];                     // 1 KB

    const int tid  = threadIdx.x;
    const int lane = tid & 31;
    const int wave = tid >> 5;
    const long long rowBase = (long long)blockIdx.x * ROWS_PER_BLOCK;

    // ---- Stage valuations tile into LDS (coalesced b128 loads), clamp rows for tail block.
#pragma unroll
    for (int i = 0; i < 8; ++i) {
        const int f   = (i * 256 + tid) * 4;   // flat float index into 128x64 tile
        const int r   = f >> 6;
        const int col = f & 63;
        long long grow = rowBase + r;
        long long last = (long long)batch - 1;
        if (grow > last) grow = last;
        v4f d = *(const v4f*)(val + grow * NITEMS + col);
        v2f t0; t0.x = d.x; t0.y = d.y;
        v2f t1; t1.x = d.z; t1.y = d.w;
        v2f* dst = (v2f*)&sV[r * SVPITCH + col];
        dst[0] = t0;
        dst[1] = t1;
    }
    sw0[tid] = w0[tid];   // blockDim.x == NHID == 256
    __syncthreads();

    // ---- Per-wave 16-row tile. Build WMMA A fragments (16x4 f32 layout, 2 VGPRs each).
    const int m0   = wave * 16;
    const int arow = lane & 15;
    const int hi   = lane >> 4;       // 0: K pair {0,1}, 1: K pair {2,3}
    const int koff = hi * 2;

    v2f afrag[16];
#pragma unroll
    for (int c = 0; c < 16; ++c)
        afrag[c] = *(const v2f*)&sV[(m0 + arow) * SVPITCH + c * 4 + koff];

    // ---- Hidden layer: 16 column-tiles of 16x16, each = 16 chained f32 WMMAs (K=64).
    const int ncol = lane & 15;
    float bestv[8];
    int   besti[8];
#pragma unroll
    for (int v = 0; v < 8; ++v) { bestv[v] = -FLT_MAX; besti[v] = 0; }

    for (int nt = 0; nt < 16; ++nt) {
        const int n = nt * 16 + ncol;
        const float* wrow = W + n * NITEMS;   // B[k][n] = W[n][k]
        v8f acc = {};
#pragma unroll
        for (int c = 0; c < 16; ++c) {
            v2f b = *(const v2f*)(wrow + c * 4 + koff);
            acc = __builtin_amdgcn_wmma_f32_16x16x4_f32(
                      false, afrag[c], false, b, (short)0, acc, false, false);
        }
        const float w0n = sw0[n];
#pragma unroll
        for (int v = 0; v < 8; ++v) {
            float h = acc[v] + w0n;
            // ascending n => strict '>' keeps first-occurrence argmax within a lane
            if (h > bestv[v]) { bestv[v] = h; besti[v] = n; }
        }
    }

    // ---- Row max/argmax: reduce across the 16 lanes of each half-wave
    // (VGPR v holds row v in lanes 0-15, row v+8 in lanes 16-31).
#pragma unroll
    for (int v = 0; v < 8; ++v) {
        float mv = bestv[v];
        int   mi = besti[v];
#pragma unroll
        for (int off = 8; off; off >>= 1) {
            float ov = __shfl_xor(mv, off, 32);
            int   oi = __shfl_xor(mi, off, 32);
            if (ov > mv || (ov == mv && oi < mi)) { mv = ov; mi = oi; }
        }
        bestv[v] = mv;
        besti[v] = mi;
    }

    // ---- Output: per row gather clamped W row, alloc store (b64/lane, coalesced),
    //      payment = dot(alloc, valuations) - relu(maxh), clamped at 0.
#pragma unroll
    for (int r = 0; r < 16; ++r) {
        const int src = (r >> 3) << 4;             // 0 for rows 0-7, 16 for rows 8-15
        const float mh = __shfl(bestv[r & 7], src, 32);
        const int   mi = __shfl(besti[r & 7], src, 32);
        const long long grow = rowBase + m0 + r;

        float a0 = 0.f, a1 = 0.f;
        if (mh > 0.f) {                            // uniform across the wave
            const float* wr = W + mi * NITEMS + lane * 2;
            a0 = fminf(fmaxf(wr[0], 0.f), 1.f);
            a1 = fminf(fmaxf(wr[1], 0.f), 1.f);
        }
        const float* vr = &sV[(m0 + r) * SVPITCH + lane * 2];
        float p = a0 * vr[0] + a1 * vr[1];
#pragma unroll
        for (int off = 16; off; off >>= 1) p += __shfl_xor(p, off, 32);
        p = fmaxf(p - fmaxf(mh, 0.f), 0.f);

        if (grow < (long long)batch) {
            v2f a; a.x = a0; a.y = a1;
            *(v2f*)&alloc_out[grow * NITEMS + lane * 2] = a;
            if (lane == 0) pay_out[grow] = p;
        }
    }
}

extern "C" void kernel_launch(void* const* d_in, const int* in_sizes, int n_in,
                              void* d_out, int out_size, void* d_ws, size_t ws_size,
                              hipStream_t stream) {
    const float* val = (const float*)d_in[0];
    const float* W   = (const float*)d_in[1];
    const float* w0  = (const float*)d_in[2];
    const int batch  = in_sizes[0] / NITEMS;

    float* alloc_out = (float*)d_out;
    float* pay_out   = alloc_out + (size_t)batch * NITEMS;

    const int grid = (batch + ROWS_PER_BLOCK - 1) / ROWS_PER_BLOCK;
    rochet_wmma_kernel<<<grid, 256, 0, stream>>>(val, W, w0, alloc_out, pay_out, batch);
}